// QueryAndGroup_6932077216284
// MI455X (gfx1250) — compile-verified
//
#include <hip/hip_runtime.h>

// ---------------------------------------------------------------------------
// Fused PointNet++ ball-query + grouping for MI455X (gfx1250, wave32).
//
// Squared distances via V_WMMA_F32_16X16X4_F32 with C = 0 (inline-hoisted):
//   A[m][:] = (-2qx, -2qy, -2qz, |q|^2)   (16x4)
//   B[:][n] = ( cx ,  cy ,  cz ,   1  )   (4x16)
//   => D[m][n] = |q_m|^2 - 2 q_m.c_n ;  in-ball test: D < r^2 - |c_n|^2
// (|c|^2 folded into the per-column comparison threshold -> no per-chunk
//  C-broadcast movs on the WMMA critical path.)
//
// Structure:
//  * Block = 4 waves = one 16-query tile. Each wave scans a disjoint quarter
//    of N=16384 candidates IN INDEX ORDER into its own LDS hit list; segment
//    order == index order, so an ordered merge reproduces "first 32 within
//    radius" exactly (pad with first hit / 0).
//  * Inner loop software-pipelined in groups of 8 chunks: next group's 8
//    global_load_b96 in flight while current group's 8 WMMAs + ballots run;
//    global_prefetch_b8 covers group g+2.
//  * Ballot of D-VGPR j gives in-ball masks for rows j (lo16) / j+8 (hi16).
//    Mask routing + append skipped entirely on hit-free tiles (~35%).
//  * Gather phase: 4 rows/wave, lane == sample slot -> 128B coalesced stores
//    for the 70MB output; 8.4MB feature tensor is L2-resident for gathers.
// ---------------------------------------------------------------------------

typedef float v2f __attribute__((ext_vector_type(2)));
typedef float v8f __attribute__((ext_vector_type(8)));

#define RADIUS2  0.01f
#define NSAMPLE  32
#define BQ       2
#define NPTS     16384
#define PQ       4096
#define CF       64
#define OCH      (3 + CF)        // 67 output channels
#define NSEG     4               // waves per block == N segments
#define SEGLEN   (NPTS / NSEG)   // 4096 candidates per wave
#define GRP      8               // chunks (of 16 candidates) per pipelined group
#define NCHUNK   (SEGLEN / 16)   // 256
#define NGRP     (NCHUNK / GRP)  // 32
#define ROWS_PER_WAVE (16 / NSEG)

__global__ __launch_bounds__(NSEG * 32)
void ballquery_group_wmma(const float* __restrict__ xyz,      // (B, N, 3)
                          const float* __restrict__ new_xyz,  // (B, P, 3)
                          const float* __restrict__ feat,     // (B, C, N)
                          float* __restrict__ out)            // (B, 67, P, 32)
{
    __shared__ int s_seg[NSEG][16][NSAMPLE];  // per-segment ordered hit lists (8KB)
    __shared__ int s_cnt[NSEG][16];           // per-segment hit counts
    __shared__ int s_idx[16][NSAMPLE];        // merged final indices (2KB)

    const int lane = threadIdx.x & 31;
    const int wv   = threadIdx.x >> 5;        // segment / wave id 0..3
    const int tile = blockIdx.x;              // 0 .. B*P/16 - 1
    const int b    = tile / (PQ / 16);
    const int q0   = (tile % (PQ / 16)) * 16;
    const int m    = lane & 15;               // row (query) this lane represents

    // ---- Query points (all 4 waves load the same 16 queries) ---------------
    const float* qp = new_xyz + ((size_t)b * PQ + q0 + m) * 3;
    const float qx = qp[0], qy = qp[1], qz = qp[2];
    const float qq = qx * qx + qy * qy + qz * qz;

    // A matrix (16x4 f32): lanes 0-15 -> K={0,1}, lanes 16-31 -> K={2,3}
    v2f A;
    if (lane < 16) { A.x = -2.0f * qx; A.y = -2.0f * qy; }
    else           { A.x = -2.0f * qz; A.y = qq;        }

    const v8f CZ = { 0.f, 0.f, 0.f, 0.f, 0.f, 0.f, 0.f, 0.f };  // hoisted out of loop

    const float* xb      = xyz + (size_t)b * NPTS * 3;
    const int    segbase = wv * SEGLEN;
    const int    col     = lane & 15;
    int          cnt     = 0;   // per-row ordered hits in this segment (lanes 0..15)

    // ---- Software-pipelined candidate scan ---------------------------------
    float cx0[GRP], cy0[GRP], cz0[GRP];
#pragma unroll
    for (int u = 0; u < GRP; ++u) {
        const float* cp = xb + (size_t)(segbase + u * 16 + col) * 3;
        cx0[u] = cp[0]; cy0[u] = cp[1]; cz0[u] = cp[2];
    }

    for (int g = 0; g < NGRP; ++g) {
        // Issue next group's loads early (8 in flight) + prefetch group g+2.
        float cx1[GRP], cy1[GRP], cz1[GRP];
        const bool have_next = (g + 1 < NGRP);
        if (have_next) {
            const int nb = segbase + (g + 1) * GRP * 16;
            __builtin_prefetch(xb + (size_t)(nb + GRP * 16) * 3, 0, 1);
#pragma unroll
            for (int u = 0; u < GRP; ++u) {
                const float* cp = xb + (size_t)(nb + u * 16 + col) * 3;
                cx1[u] = cp[0]; cy1[u] = cp[1]; cz1[u] = cp[2];
            }
        }

        // Process current group: 8 WMMA tiles of 16 queries x 16 candidates.
#pragma unroll
        for (int u = 0; u < GRP; ++u) {
            const float cx = cx0[u], cy = cy0[u], cz = cz0[u];
            // Per-column threshold: d^2 < r^2  <=>  D < r^2 - |c|^2
            const float thr = RADIUS2 - (cx * cx + cy * cy + cz * cz);

            v2f Bm;
            if (lane < 16) { Bm.x = cx; Bm.y = cy;   }
            else           { Bm.x = cz; Bm.y = 1.0f; }

            v8f D = __builtin_amdgcn_wmma_f32_16x16x4_f32(
                false, A, false, Bm, (short)0, CZ, false, false);

            unsigned bal[8];
#pragma unroll
            for (int j = 0; j < 8; ++j)
                bal[j] = __builtin_amdgcn_ballot_w32(D[j] < thr);

            // Skip routing/append entirely when the whole 16x16 tile is empty
            // (~35% of tiles). 'any' is wave-uniform -> scalar branch.
            const unsigned any = bal[0] | bal[1] | bal[2] | bal[3] |
                                 bal[4] | bal[5] | bal[6] | bal[7];
            if (any) {
                // Route row m's 16-bit mask to lane m.
                unsigned bm = bal[0];
                bm = ((m & 7) == 1) ? bal[1] : bm;
                bm = ((m & 7) == 2) ? bal[2] : bm;
                bm = ((m & 7) == 3) ? bal[3] : bm;
                bm = ((m & 7) == 4) ? bal[4] : bm;
                bm = ((m & 7) == 5) ? bal[5] : bm;
                bm = ((m & 7) == 6) ? bal[6] : bm;
                bm = ((m & 7) == 7) ? bal[7] : bm;
                unsigned rowmask = (bm >> ((m >> 3) ? 16 : 0)) & 0xFFFFu;

                if (lane < 16) {
                    const int cbase = segbase + (g * GRP + u) * 16;
                    while (rowmask && cnt < NSAMPLE) {
                        const int bit = __builtin_ctz(rowmask);
                        rowmask &= rowmask - 1;
                        s_seg[wv][m][cnt] = cbase + bit;
                        ++cnt;
                    }
                }
            }
        }

        if (have_next) {
#pragma unroll
            for (int u = 0; u < GRP; ++u) {
                cx0[u] = cx1[u]; cy0[u] = cy1[u]; cz0[u] = cz1[u];
            }
        }

        // Early exit once every row of this wave's segment list is full.
        const bool done = (lane < 16) ? (cnt >= NSAMPLE) : true;
        if (__builtin_amdgcn_ballot_w32(done) == 0xFFFFFFFFu) break;
    }

    if (lane < 16) s_cnt[wv][m] = cnt;
    __syncthreads();

    // ---- Ordered merge of the 4 segment lists (16 lanes, one row each) -----
    if (threadIdx.x < 16) {
        const int row = threadIdx.x;
        int total = 0;
        for (int seg = 0; seg < NSEG && total < NSAMPLE; ++seg) {
            const int c = s_cnt[seg][row];
            for (int k = 0; k < c && total < NSAMPLE; ++k)
                s_idx[row][total++] = s_seg[seg][row][k];
        }
        const int first = (total > 0) ? s_idx[row][0] : 0;
        for (int k = total; k < NSAMPLE; ++k) s_idx[row][k] = first;
    }
    __syncthreads();

    // =======================================================================
    // Grouping phase: wave wv handles rows [wv*4, wv*4+4). lane == sample s.
    // Output stores: 32 consecutive floats (128B) -> fully coalesced.
    // =======================================================================
    const size_t chstride = (size_t)PQ * NSAMPLE;
    const float* fb = feat + (size_t)b * CF * NPTS;

    for (int rr = 0; rr < ROWS_PER_WAVE; ++rr) {
        const int r = wv * ROWS_PER_WAVE + rr;
        const int p = q0 + r;
        const int i = s_idx[r][lane];

        // Query center for row r: broadcast from lane r of this wave.
        const float nqx = __shfl(qx, r, 32);
        const float nqy = __shfl(qy, r, 32);
        const float nqz = __shfl(qz, r, 32);

        const float* pc = xb + (size_t)i * 3;
        const float gx = pc[0] - nqx;
        const float gy = pc[1] - nqy;
        const float gz = pc[2] - nqz;

        size_t ob = (size_t)b * OCH * chstride + (size_t)p * NSAMPLE + lane;
        out[ob + 0 * chstride] = gx;
        out[ob + 1 * chstride] = gy;
        out[ob + 2 * chstride] = gz;

        ob += 3 * chstride;
#pragma unroll 8
        for (int c = 0; c < CF; ++c) {
            out[ob + (size_t)c * chstride] = fb[(size_t)c * NPTS + i];
        }
    }
}

extern "C" void kernel_launch(void* const* d_in, const int* in_sizes, int n_in,
                              void* d_out, int out_size, void* d_ws, size_t ws_size,
                              hipStream_t stream) {
    (void)in_sizes; (void)n_in; (void)out_size; (void)d_ws; (void)ws_size;
    const float* xyz     = (const float*)d_in[0];  // (B, N, 3)
    const float* new_xyz = (const float*)d_in[1];  // (B, P, 3)
    const float* feat    = (const float*)d_in[2];  // (B, C, N)
    float*       out     = (float*)d_out;          // (B, 67, P, 32)

    const int nblocks = BQ * (PQ / 16);            // 512 tiles, 4 waves each
    ballquery_group_wmma<<<nblocks, NSEG * 32, 0, stream>>>(xyz, new_xyz, feat, out);
}